// Embedding_wlogits_28887950033164
// MI455X (gfx1250) — compile-verified
//
#include <hip/hip_runtime.h>
#include <stdint.h>

// Problem constants from the reference: x[4,1024,32000] f32, emb[32000,512] f32.
#define VOCAB   32000
#define DMODEL  512
#define NTH     256
#define NV4     (VOCAB / 4)      // 8000 float4 per row
#define CHUNK   1024             // float4 per LDS chunk (16 KB)
#define NCHUNK  8                // ceil(8000 / 1024)
#define IPT     (CHUNK / NTH)    // 4 async-load issues per thread per chunk

// CPol temporal hint: TH[2:0] = 1 -> non-temporal (streamed-once data).
#define CPOL_NT 1

#if defined(__HIP_DEVICE_COMPILE__) && defined(__gfx1250__) &&               \
    __has_builtin(__builtin_amdgcn_global_load_async_to_lds_b128) &&         \
    __has_builtin(__builtin_amdgcn_s_wait_asynccnt)
#define USE_ASYNC 1
#else
#define USE_ASYNC 0
#endif

// Native clang vector for nontemporal builtins (HIP float2 is a class type).
typedef float f2v __attribute__((ext_vector_type(2)));

#if USE_ASYNC
// Exact parameter type per clang diagnostic: vector_size(16) int, AS(1)/AS(3).
typedef int v4i __attribute__((vector_size(16)));
typedef __attribute__((address_space(1))) v4i gv4i_t;
typedef __attribute__((address_space(3))) v4i lv4i_t;

__device__ __forceinline__ gv4i_t* to_gmem(const void* p) {
    return (gv4i_t*)(uintptr_t)p;            // flat global == AS(1) bitwise
}
__device__ __forceinline__ lv4i_t* to_lds(void* p) {
    return (lv4i_t*)(uint32_t)(uintptr_t)p;  // low 32 bits of flat LDS = offset
}
#endif

__global__ __launch_bounds__(NTH) void argmax_gather_kernel(
    const float* __restrict__ x, const float* __restrict__ emb,
    float* __restrict__ out) {
    __shared__ float4 sbuf[2][CHUNK];          // 32 KB double buffer
    __shared__ float  swv[NTH / 32];
    __shared__ int    swi[NTH / 32];
    __shared__ int    sbest;

    const int t   = threadIdx.x;
    const int row = blockIdx.x;
    const float4* xrow = (const float4*)(x + (size_t)row * VOCAB);

    float bv = -__builtin_huge_valf();
    int   bi = 0;

#if USE_ASYNC
    // Prologue: issue chunk 0 into buffer 0 (4 async b128 loads per thread).
    {
#pragma unroll
        for (int r = 0; r < IPT; ++r) {
            int l = r * NTH + t;
            int g = l; if (g > NV4 - 1) g = NV4 - 1;   // clamp tail (dup, skipped later)
            __builtin_amdgcn_global_load_async_to_lds_b128(
                to_gmem(xrow + g), to_lds(&sbuf[0][l]), 0, CPOL_NT);
        }
    }
    for (int c = 0; c < NCHUNK; ++c) {
        if (c + 1 < NCHUNK) {
            // Issue next chunk into the other buffer, then retire current chunk.
#pragma unroll
            for (int r = 0; r < IPT; ++r) {
                int l = r * NTH + t;
                int g = (c + 1) * CHUNK + l; if (g > NV4 - 1) g = NV4 - 1;
                __builtin_amdgcn_global_load_async_to_lds_b128(
                    to_gmem(xrow + g), to_lds(&sbuf[(c + 1) & 1][l]), 0, CPOL_NT);
            }
            __builtin_amdgcn_s_wait_asynccnt(IPT);  // chunk c retired; c+1 in flight
        } else {
            __builtin_amdgcn_s_wait_asynccnt(0);
        }
        __syncthreads();  // all waves' async writes for chunk c visible
        const float4* cb = (const float4*)sbuf[c & 1];
#pragma unroll
        for (int r = 0; r < IPT; ++r) {
            int l = r * NTH + t;
            int g = c * CHUNK + l;
            if (g < NV4) {
                float4 v = cb[l];
                int base = g * 4;
                if (v.x > bv) { bv = v.x; bi = base; }
                if (v.y > bv) { bv = v.y; bi = base + 1; }
                if (v.z > bv) { bv = v.z; bi = base + 2; }
                if (v.w > bv) { bv = v.w; bi = base + 3; }
            }
        }
        __syncthreads();  // buffer may be re-filled next iteration
    }
#else
    // Fallback: direct vectorized streaming loads.
    for (int g = t; g < NV4; g += NTH) {
        float4 v = xrow[g];
        int base = g * 4;
        if (v.x > bv) { bv = v.x; bi = base; }
        if (v.y > bv) { bv = v.y; bi = base + 1; }
        if (v.z > bv) { bv = v.z; bi = base + 2; }
        if (v.w > bv) { bv = v.w; bi = base + 3; }
    }
#endif

    // Wave32 reduction (first-index tie-break, matching jnp.argmax).
#pragma unroll
    for (int off = 16; off > 0; off >>= 1) {
        float ov = __shfl_down(bv, off, 32);
        int   oi = __shfl_down(bi, off, 32);
        if (ov > bv || (ov == bv && oi < bi)) { bv = ov; bi = oi; }
    }
    if ((t & 31) == 0) { swv[t >> 5] = bv; swi[t >> 5] = bi; }
    __syncthreads();
    if (t == 0) {
        float v = swv[0]; int i = swi[0];
#pragma unroll
        for (int w = 1; w < NTH / 32; ++w) {
            if (swv[w] > v || (swv[w] == v && swi[w] < i)) { v = swv[w]; i = swi[w]; }
        }
        sbest = i;
    }
    __syncthreads();

    // Gather the winning embedding row: 256 threads x float2 = 512 floats.
    // Embedding loads keep default (RT) policy so rows get L2 reuse across blocks;
    // output is written once and never re-read -> non-temporal store.
    const f2v* erow = (const f2v*)(emb + (size_t)sbest * DMODEL);
    f2v*       orow = (f2v*)(out + (size_t)row * DMODEL);
    f2v val = erow[t];
    __builtin_nontemporal_store(val, &orow[t]);
}

extern "C" void kernel_launch(void* const* d_in, const int* in_sizes, int n_in,
                              void* d_out, int out_size, void* d_ws, size_t ws_size,
                              hipStream_t stream) {
    const float* x   = (const float*)d_in[0];   // [4,1024,32000] f32
    const float* emb = (const float*)d_in[1];   // [32000,512] f32
    float*       out = (float*)d_out;           // [4,1024,512] f32
    (void)n_in; (void)out_size; (void)d_ws; (void)ws_size;

    const int rows = in_sizes[0] / VOCAB;       // 4096
    hipLaunchKernelGGL(argmax_gather_kernel, dim3(rows), dim3(NTH), 0, stream,
                       x, emb, out);
}